// LSAGEDirected_67336497266901
// MI455X (gfx1250) — compile-verified
//
#include <hip/hip_runtime.h>
#include <hip/hip_bf16.h>

// LSAGEDirected on MI455X (gfx1250, wave32).
//   - 3 scatter-aggregation hops (L2-atomic bound; all h buffers fit in 192MB L2)
//   - final [50000,512]x[512,128] projection on the matrix engine via
//     V_WMMA_F32_16X16X4_F32 (full fp32 precision, matches fp32 reference).

typedef __attribute__((ext_vector_type(2))) float v2f;
typedef __attribute__((ext_vector_type(8))) float v8f;

#define N_NODES 50000
#define N_EDGES 800000
#define IN_DIM 64
#define OUT_DIM 128
#define FINAL_DIM 512

// ---------------- utility kernels ----------------

__global__ void zero_f32(float* __restrict__ p, long long n) {
    long long i = (long long)blockIdx.x * blockDim.x + threadIdx.x;
    long long stride = (long long)gridDim.x * blockDim.x;
    for (; i < n; i += stride) p[i] = 0.0f;
}

__global__ void degree_kernel(const int* __restrict__ src, const int* __restrict__ dst,
                              float* __restrict__ deg_in, float* __restrict__ deg_out, int e_cnt) {
    int e = blockIdx.x * blockDim.x + threadIdx.x;
    if (e < e_cnt) {
        atomicAdd(&deg_in[dst[e]], 1.0f);
        atomicAdd(&deg_out[src[e]], 1.0f);
    }
}

__global__ void invert_kernel(float* __restrict__ p, int n) {
    int i = blockIdx.x * blockDim.x + threadIdx.x;
    if (i < n) p[i] = 1.0f / fmaxf(p[i], 1.0f);
}

// Scatter-add one hop: for edge (s,t) add h[s] into hn[t][0:d] (in-half)
// and h[t] into hn[s][d:2d] (out-half). One thread = one edge x 4 features.
__global__ void agg_kernel(const float* __restrict__ h, const int* __restrict__ src,
                           const int* __restrict__ dst, float* __restrict__ hn,
                           int d, int e_cnt) {
    const int chunks = d >> 2;  // float4 chunks per row
    long long tid = (long long)blockIdx.x * blockDim.x + threadIdx.x;
    long long total = (long long)e_cnt * chunks;
    if (tid >= total) return;
    int e  = (int)(tid / chunks);
    int c4 = (int)(tid % chunks) << 2;
    int s = src[e], t = dst[e];

    const float4 hs = *(const float4*)(h + (size_t)s * d + c4);
    const float4 ht = *(const float4*)(h + (size_t)t * d + c4);

    float* oi = hn + (size_t)t * (2 * d) + c4;        // in-aggregation -> dst row, first half
    float* oo = hn + (size_t)s * (2 * d) + d + c4;    // out-aggregation -> src row, second half
    atomicAdd(oi + 0, hs.x); atomicAdd(oi + 1, hs.y);
    atomicAdd(oi + 2, hs.z); atomicAdd(oi + 3, hs.w);
    atomicAdd(oo + 0, ht.x); atomicAdd(oo + 1, ht.y);
    atomicAdd(oo + 2, ht.z); atomicAdd(oo + 3, ht.w);
}

// Normalize: first half of each row by inv_in[row], second half by inv_out[row].
__global__ void scale_kernel(float* __restrict__ hn, const float* __restrict__ inv_in,
                             const float* __restrict__ inv_out, int d, int n_nodes) {
    long long tid = (long long)blockIdx.x * blockDim.x + threadIdx.x;
    long long total = (long long)n_nodes * (2 * d);
    if (tid >= total) return;
    int row = (int)(tid / (2 * d));
    int col = (int)(tid % (2 * d));
    hn[tid] *= (col < d) ? inv_in[row] : inv_out[row];
}

// ---------------- WMMA projection ----------------
// out[N,128] = h3[N,512] @ W[128,512]^T + bias
// One wave per 16-row M tile; each wave accumulates all 8 N-tiles (reuses the
// A fetch 8x). 512/4 = 128 k-steps, 8 WMMAs each => 1024 v_wmma per wave.
//
// f32 WMMA operand layouts (ISA 7.12.2):
//   A 16x4 : lane = half*16 + m ; v[0]=A[m][2*half], v[1]=A[m][2*half+1]
//   B 4x16 : lane = half*16 + n ; v[0]=B[2*half][n], v[1]=B[2*half+1][n]
//   C 16x16: lane = half*16 + n ; v[r]=C[8*half + r][n]
__global__ __launch_bounds__(256) void gemm_wmma_kernel(const float* __restrict__ h3,
                                                        const float* __restrict__ Wm,
                                                        const float* __restrict__ bias,
                                                        float* __restrict__ out,
                                                        int n_nodes) {
    const int wave = blockIdx.x * (blockDim.x >> 5) + (threadIdx.x >> 5);
    const int mt = wave;                       // M tile index
    if (mt * 16 >= n_nodes) return;            // wave-uniform exit; EXEC stays full
    const int lane = threadIdx.x & 31;
    const int half = lane >> 4;                // 0: lanes 0-15, 1: lanes 16-31
    const int l15  = lane & 15;

    const int m = mt * 16 + l15;               // A-matrix row for this lane
    const float* arow = h3 + (size_t)m * FINAL_DIM + half * 2;

    v8f acc[8];
#pragma unroll
    for (int t = 0; t < 8; ++t) acc[t] = (v8f){0.f, 0.f, 0.f, 0.f, 0.f, 0.f, 0.f, 0.f};

    for (int k0 = 0; k0 < FINAL_DIM; k0 += 4) {
        v2f a;
        a[0] = arow[k0];
        a[1] = arow[k0 + 1];
#pragma unroll
        for (int t = 0; t < 8; ++t) {
            const int n = t * 16 + l15;        // output column
            // B[k][n] = W[n][k] (W is row-major [128,512])
            const float* wrow = Wm + (size_t)n * FINAL_DIM + k0 + half * 2;
            v2f bv;
            bv[0] = wrow[0];
            bv[1] = wrow[1];
            acc[t] = __builtin_amdgcn_wmma_f32_16x16x4_f32(
                /*neg_a=*/false, a, /*neg_b=*/false, bv,
                /*c_mod=*/(short)0, acc[t], /*reuse_a=*/false, /*reuse_b=*/false);
        }
    }

#pragma unroll
    for (int t = 0; t < 8; ++t) {
        const int col = t * 16 + l15;
        const float bc = bias[col];
#pragma unroll
        for (int r = 0; r < 8; ++r) {
            const int row = mt * 16 + 8 * half + r;
            out[(size_t)row * OUT_DIM + col] = acc[t][r] + bc;
        }
    }
}

// ---------------- launcher ----------------

extern "C" void kernel_launch(void* const* d_in, const int* in_sizes, int n_in,
                              void* d_out, int out_size, void* d_ws, size_t ws_size,
                              hipStream_t stream) {
    const float* feature = (const float*)d_in[0];   // [N, 64]
    const int*   edge    = (const int*)d_in[1];     // [2, E] (int32 on device)
    const float* Wm      = (const float*)d_in[2];   // [128, 512]
    const float* bias    = (const float*)d_in[3];   // [128]
    float*       out     = (float*)d_out;           // [N, 128]

    const int N = N_NODES;
    const int E = N_EDGES;
    const int* src = edge;
    const int* dst = edge + E;

    // Workspace layout (floats): inv_in[N] | inv_out[N] | h1[N*128] | h2[N*256] | h3[N*512]
    float* ws     = (float*)d_ws;
    float* inv_in  = ws;
    float* inv_out = ws + N;
    float* h1 = inv_out + N;
    float* h2 = h1 + (size_t)N * 128;
    float* h3 = h2 + (size_t)N * 256;
    const long long ws_floats = (long long)N * (2 + 128 + 256 + 512);

    const int B = 256;

    // 1) zero degree accumulators + hop buffers (ws is poisoned by harness)
    zero_f32<<<2048, B, 0, stream>>>(ws, ws_floats);

    // 2) degrees (accumulated into inv_in/inv_out, inverted in place)
    degree_kernel<<<(E + B - 1) / B, B, 0, stream>>>(src, dst, inv_in, inv_out, E);
    invert_kernel<<<(2 * N + B - 1) / B, B, 0, stream>>>(inv_in, 2 * N);

    // 3) hop 0: feature(64) -> h1(128)
    {
        long long total = (long long)E * (IN_DIM / 4);
        agg_kernel<<<(int)((total + B - 1) / B), B, 0, stream>>>(feature, src, dst, h1, IN_DIM, E);
        long long st = (long long)N * 128;
        scale_kernel<<<(int)((st + B - 1) / B), B, 0, stream>>>(h1, inv_in, inv_out, IN_DIM, N);
    }
    // 4) hop 1: h1(128) -> h2(256)
    {
        long long total = (long long)E * (128 / 4);
        agg_kernel<<<(int)((total + B - 1) / B), B, 0, stream>>>(h1, src, dst, h2, 128, E);
        long long st = (long long)N * 256;
        scale_kernel<<<(int)((st + B - 1) / B), B, 0, stream>>>(h2, inv_in, inv_out, 128, N);
    }
    // 5) hop 2: h2(256) -> h3(512)
    {
        long long total = (long long)E * (256 / 4);
        agg_kernel<<<(int)((total + B - 1) / B), B, 0, stream>>>(h2, src, dst, h3, 256, E);
        long long st = (long long)N * 512;
        scale_kernel<<<(int)((st + B - 1) / B), B, 0, stream>>>(h3, inv_in, inv_out, 256, N);
    }

    // 6) projection on the matrix engine: one wave per 16-row tile (50000/16 = 3125 waves)
    {
        const int waves = N / 16;                 // exact
        const int wpb = 256 / 32;                 // 8 waves per block
        const int grid = (waves + wpb - 1) / wpb; // 391
        gemm_wmma_kernel<<<grid, 256, 0, stream>>>(h3, Wm, bias, out, N);
    }
}